// blackout4_3599182594545
// MI455X (gfx1250) — compile-verified
//
#include <hip/hip_runtime.h>

#define EPS 1e-10f

typedef __attribute__((ext_vector_type(2))) float v2f;
typedef __attribute__((ext_vector_type(8))) float v8f;

// ---------------------------------------------------------------------------
// Phase 1: one workgroup per row. Stream the 32000-float row with b128 loads
// (uniform scalar trip count => no per-lane predication in the hot loop),
// keep running max in VGPRs, wave32 shfl reduce + 8-entry LDS reduce, then
// thread 0 computes the 6-term BlackOut loss contribution for this row.
// Pure HBM-bandwidth bound: 1.049 GB once => ~45 us floor at 23.3 TB/s.
// ---------------------------------------------------------------------------
__global__ __launch_bounds__(256) void blackout_rowloss_kernel(
    const float* __restrict__ yHat,
    const float* __restrict__ prob,
    const int*   __restrict__ y,
    const int*   __restrict__ ind,
    float*       __restrict__ partial,
    int V, int S, int K)
{
  const int b   = blockIdx.x;
  const int tid = threadIdx.x;
  const float* __restrict__ row = yHat + (size_t)b * (size_t)V;

  const int V4 = V >> 2;
  const float4* __restrict__ row4 = reinterpret_cast<const float4*>(row);

  float m = -3.402823466e+38f;

  // Main body: nk full sweeps of 256 lanes x float4. Trip count is scalar =>
  // s_cmp/s_cbranch control, unconditional global_load_b128 in the body.
  const int nk = V4 >> 8;                      // V4 / 256
  #pragma unroll 4
  for (int k = 0; k < nk; ++k) {
    float4 v = row4[(k << 8) + tid];
    m = fmaxf(m, fmaxf(fmaxf(v.x, v.y), fmaxf(v.z, v.w)));
  }
  // Remainder float4s (V4 % 256 lanes active once)
  {
    const int ri = (nk << 8) + tid;
    if (ri < V4) {
      float4 v = row4[ri];
      m = fmaxf(m, fmaxf(fmaxf(v.x, v.y), fmaxf(v.z, v.w)));
    }
  }
  // Scalar tail for V % 4 (empty for V=32000)
  for (int i = (V4 << 2) + tid; i < V; i += 256) m = fmaxf(m, row[i]);

  // wave32 butterfly reduce
  #pragma unroll
  for (int off = 16; off > 0; off >>= 1)
    m = fmaxf(m, __shfl_xor(m, off, 32));

  __shared__ float smax[8];
  if ((tid & 31) == 0) smax[tid >> 5] = m;
  __syncthreads();

  if (tid == 0) {
    float mm = smax[0];
    #pragma unroll
    for (int w = 1; w < 8; ++w) mm = fmaxf(mm, smax[w]);

    // ---- tiny epilogue (K=5): gathers hit L2/WGP$ (row was just read) ----
    const int yy = y[b];
    const float yg = row[yy] - mm;                       // target logit, shifted
    const float* __restrict__ prow = prob + (size_t)yy * (size_t)S;

    float p[8];
    float c[8];
    float q = 3.402823466e+38f;
    for (int k = 0; k < K; ++k) {
      const int id = ind[(size_t)b * (size_t)K + k];     // id in [0, S)
      c[k] = row[id] - mm;                               // reference gathers yh[:, id]
      p[k] = 1.0f / prow[id];
      q = fminf(q, p[k]);
    }

    float out0 = q * expf(yg);
    float sum  = out0;
    float outk[8];
    for (int k = 0; k < K; ++k) { outk[k] = p[k] * expf(c[k]); sum += outk[k]; }
    const float inv = 1.0f / sum;

    // mask picks log(out) at slot 0, log(1-out) at slots 1..K
    float s = logf(out0 * inv + EPS);
    for (int k = 0; k < K; ++k) s += logf(1.0f - outk[k] * inv + EPS);

    partial[b] = s;
  }
}

// ---------------------------------------------------------------------------
// Phase 2: sum the B per-row partials on the matrix pipe.
// D = ones(16x4) x B + C : every row of D holds the 16 column sums of the
// 4x16 B operand; which memory element lands in which B slot is irrelevant
// for a total sum, so one b128 load per lane feeds two WMMAs (128 values /
// iteration) with zero predication in the body. Single wave32, EXEC all ones
// at every WMMA, fixed order => deterministic.
// ---------------------------------------------------------------------------
__global__ __launch_bounds__(32) void blackout_reduce_wmma_kernel(
    const float* __restrict__ partial,
    float*       __restrict__ out,
    int n, float neg_inv_count)
{
  const int lane = threadIdx.x;

  v2f ones; ones[0] = 1.0f; ones[1] = 1.0f;  // A = 16x4 all-ones (2 VGPRs/lane)
  v8f acc = {};                              // C/D = 16x16 f32 accumulator

  const int nbody = n & ~127;                // 128 values per iteration
  const float4* __restrict__ p4 = reinterpret_cast<const float4*>(partial);
  for (int base = 0; base < nbody; base += 128) {
    float4 t = p4[(base >> 2) + lane];       // lane holds 4 consecutive floats
    v2f b0; b0[0] = t.x; b0[1] = t.y;
    v2f b1; b1[0] = t.z; b1[1] = t.w;
    acc = __builtin_amdgcn_wmma_f32_16x16x4_f32(
        false, ones, false, b0, (short)0, acc, false, false);
    acc = __builtin_amdgcn_wmma_f32_16x16x4_f32(
        false, ones, false, b1, (short)0, acc, false, false);
  }

  // Tail (n % 128 values), uniform outer guard, masked per-lane fills.
  if (nbody < n) {
    const int i0 = nbody + lane * 4;
    v2f b0, b1;
    b0[0] = (i0     < n) ? partial[i0]     : 0.0f;
    b0[1] = (i0 + 1 < n) ? partial[i0 + 1] : 0.0f;
    b1[0] = (i0 + 2 < n) ? partial[i0 + 2] : 0.0f;
    b1[1] = (i0 + 3 < n) ? partial[i0 + 3] : 0.0f;
    acc = __builtin_amdgcn_wmma_f32_16x16x4_f32(
        false, ones, false, b0, (short)0, acc, false, false);
    acc = __builtin_amdgcn_wmma_f32_16x16x4_f32(
        false, ones, false, b1, (short)0, acc, false, false);
  }

  // Row 0 of D lives in acc[0] on lanes 0..15 (lanes 16..31 duplicate row 8).
  // Butterfly-sum within each 16-lane half; lane 0 ends with the exact total.
  float v = acc[0];
  #pragma unroll
  for (int off = 8; off > 0; off >>= 1)
    v += __shfl_xor(v, off, 32);

  if (lane == 0) out[0] = v * neg_inv_count;   // loss = -mean
}

// ---------------------------------------------------------------------------
extern "C" void kernel_launch(void* const* d_in, const int* in_sizes, int n_in,
                              void* d_out, int out_size, void* d_ws, size_t ws_size,
                              hipStream_t stream) {
  const float* yHat = (const float*)d_in[0];   // [B, V]
  const float* prob = (const float*)d_in[1];   // [V, S]
  const int*   y    = (const int*)d_in[2];     // [B]
  const int*   ind  = (const int*)d_in[3];     // [B, K]

  const int B = in_sizes[2];
  const int V = in_sizes[0] / B;
  const int S = in_sizes[1] / V;
  const int K = in_sizes[3] / B;

  float* partial = (float*)d_ws;               // B floats (32 KB for B=8192)

  blackout_rowloss_kernel<<<B, 256, 0, stream>>>(yHat, prob, y, ind, partial, V, S, K);

  const float neg_inv_count = -1.0f / (float)((long long)B * (long long)(K + 1));
  blackout_reduce_wmma_kernel<<<1, 32, 0, stream>>>(partial, (float*)d_out, B, neg_inv_count);
}